// FactorizedSelfBilinearPooling_24550033064604
// MI455X (gfx1250) — compile-verified
//
#include <hip/hip_runtime.h>

typedef __attribute__((ext_vector_type(4))) float v4f;
typedef __attribute__((ext_vector_type(2))) float v2f;
typedef __attribute__((ext_vector_type(8))) float v8f;

#define SLICES 512          // 8 batches * 64 channels
#define PARTS 4             // blocks per slice
#define THREADS 256         // 8 wave32 per block
#define TASKS_PER_PART 4096 // (32*32*16)/PARTS quad-window tasks

// One block reduces a quarter of one (b,c) 64^3 slice.
// task -> (hp, wp, dq): window pair rows 2hp..2hp+1, cols 2wp..2wp+1,
// d-quad dq covers d = 4dq..4dq+3 (two 2-wide d windows per float4).
__global__ __launch_bounds__(THREADS)
void pool_reduce_kernel(const float* __restrict__ x, float* __restrict__ ws) {
    const int bid   = blockIdx.x;
    const int slice = bid >> 2;
    const int part  = bid & 3;
    const int t     = threadIdx.x;
    const float* xb = x + (size_t)slice * 262144u;

    float accm = 0.f;   // sum of window maxima
    float accs = 0.f;   // sum of all elements
    const int task0 = part * TASKS_PER_PART + t;
#pragma unroll 4
    for (int i = 0; i < TASKS_PER_PART / THREADS; ++i) {
        const int task = task0 + i * THREADS;
        const int hp = task >> 9;
        const int wp = (task >> 4) & 31;
        const int dq = task & 15;
        const float* p = xb + hp * 8192 + wp * 128 + dq * 4;
        // four rows of the 2x2x(d) window footprint, 16B each, streamed (NT)
        v4f r0 = __builtin_nontemporal_load((const v4f*)(p));
        v4f r1 = __builtin_nontemporal_load((const v4f*)(p + 64));
        v4f r2 = __builtin_nontemporal_load((const v4f*)(p + 4096));
        v4f r3 = __builtin_nontemporal_load((const v4f*)(p + 4160));
        // window 0 = .x/.y lanes, window 1 = .z/.w lanes
        float m0 = fmaxf(fmaxf(fmaxf(r0.x, r0.y), fmaxf(r1.x, r1.y)),
                         fmaxf(fmaxf(r2.x, r2.y), fmaxf(r3.x, r3.y)));
        float m1 = fmaxf(fmaxf(fmaxf(r0.z, r0.w), fmaxf(r1.z, r1.w)),
                         fmaxf(fmaxf(r2.z, r2.w), fmaxf(r3.z, r3.w)));
        float s0 = ((r0.x + r0.y) + (r1.x + r1.y)) + ((r2.x + r2.y) + (r3.x + r3.y));
        float s1 = ((r0.z + r0.w) + (r1.z + r1.w)) + ((r2.z + r2.w) + (r3.z + r3.w));
        accm += m0 + m1;
        accs += s0 + s1;
    }

    __shared__ float smax[THREADS];
    __shared__ float ssum[THREADS];
    smax[t] = accm;
    ssum[t] = accs;
    __syncthreads();
    for (int sft = THREADS >> 1; sft > 0; sft >>= 1) {
        if (t < sft) {
            smax[t] += smax[t + sft];
            ssum[t] += ssum[t + sft];
        }
        __syncthreads();
    }
    if (t == 0) ws[bid] = smax[0] + ssum[0] * 0.125f;  // partial s contribution
}

// Single wave32. Builds pooled P[16][64] (rows 8..15 zero), accumulates the
// 16x16 Gram P*P^T with V_WMMA_F32_16X16X4_F32 (exact fp32), takes the
// diagonal as per-batch sum(pooled^2), normalizes.
__global__ __launch_bounds__(32)
void finalize_kernel(const float* __restrict__ ws, float* __restrict__ out) {
    __shared__ float P[16 * 64];
    __shared__ float D[16 * 16];
    const int lane = threadIdx.x;

    for (int i = 0; i < 32; ++i) P[i * 32 + lane] = 0.f;
    __syncthreads();
    for (int i = 0; i < 16; ++i) {
        const int idx = i * 32 + lane;  // idx = b*64 + c, b < 8
        float s = ws[idx * 4] + ws[idx * 4 + 1] + ws[idx * 4 + 2] + ws[idx * 4 + 3];
        P[idx] = s * s;                 // pooled
    }
    __syncthreads();

    // A 16x4 fp32 layout: row M = lane&15; lane half picks K-pair, VGPR picks
    // within pair. B 4x16 mirrors it, so for the symmetric Gram a == b.
    const int m  = lane & 15;
    const int kb = (lane >> 4) * 2;
    v8f c = {};
    for (int k0 = 0; k0 < 64; k0 += 4) {
        v2f a;
        a.x = P[m * 64 + k0 + kb];
        a.y = P[m * 64 + k0 + kb + 1];
        c = __builtin_amdgcn_wmma_f32_16x16x4_f32(false, a, false, a,
                                                  (short)0, c, false, false);
    }

    // C/D layout: VGPR r holds M = r (lanes 0-15) / M = r+8 (lanes 16-31), N = lane&15
#pragma unroll
    for (int r = 0; r < 8; ++r) {
        const int mm = r + (lane >> 4) * 8;
        D[mm * 16 + (lane & 15)] = c[r];
    }
    __syncthreads();

    for (int i = 0; i < 16; ++i) {
        const int idx = i * 32 + lane;
        const int b = idx >> 6;
        const float norm = fmaxf(sqrtf(D[b * 16 + b]), 1e-12f);
        out[idx] = P[idx] / norm;
    }
}

extern "C" void kernel_launch(void* const* d_in, const int* in_sizes, int n_in,
                              void* d_out, int out_size, void* d_ws, size_t ws_size,
                              hipStream_t stream) {
    const float* x = (const float*)d_in[0];
    float* out = (float*)d_out;
    float* ws = (float*)d_ws;  // needs SLICES*PARTS floats = 8 KiB

    pool_reduce_kernel<<<SLICES * PARTS, THREADS, 0, stream>>>(x, ws);
    finalize_kernel<<<1, 32, 0, stream>>>(ws, out);
}